// GATE_33998961115547
// MI455X (gfx1250) — compile-verified
//
#include <hip/hip_runtime.h>
#include <hip/hip_bf16.h>

// ---------------------------------------------------------------------------
// GAT autoencoder forward on MI455X (gfx1250, wave32, WMMA).
//   H1 = X @ W0                      [16384 x 512]
//   H1 = gat(H1, a0_0, a1_0)         [16384 x 512]
//   H2 = H1 @ W1                     [16384 x 256]
//   H2 = gat(H2, a0_1, a1_1)         [16384 x 256]
//   A  = sigmoid(H2 @ H2^T)          [16384 x 16384]  <- dominant (137 GFLOP, 1.07GB out)
// Dense GEMMs use V_WMMA_F32_16X16X4_F32 (full fp32 precision).
// Decode GEMM: 32x64 tile/wave (8 acc chains, B reused x2 -> ~10.7 FLOP/B from
// L2), non-temporal output stores so the 16MB H2 working set stays L2-resident
// while the 1.07GB output streams to HBM.
// ---------------------------------------------------------------------------

typedef float v2f __attribute__((ext_vector_type(2)));
typedef float v8f __attribute__((ext_vector_type(8)));

#define WMMA_F32X4(a, b, c) \
    __builtin_amdgcn_wmma_f32_16x16x4_f32(false, (a), false, (b), (short)0, (c), false, false)

__device__ __forceinline__ float sigmoidf_dev(float x) {
    return 1.0f / (1.0f + __expf(-x));
}

// ---------------------------------------------------------------------------
// Zero fill
// ---------------------------------------------------------------------------
__global__ void zero_f32(float* __restrict__ p, long n) {
    long i = (long)blockIdx.x * blockDim.x + threadIdx.x;
    if (i < n) p[i] = 0.0f;
}

// ---------------------------------------------------------------------------
// C[M x N] = A[M x K] @ B[K x N], all row-major fp32.
// One wave per 16(M) x 64(N) strip; 4 accumulators reuse the A fragment and
// give 4 independent WMMA chains per K-step.
// grid = (M/16, N/64), block = 32.
// A-fragment layout (16x4 f32): lanes 0-15 carry K=0,1; lanes 16-31 carry K=2,3.
// C/D layout: VGPR r holds M=r (lanes 0-15) / M=r+8 (lanes 16-31).
// ---------------------------------------------------------------------------
__global__ void wmma_gemm_nn(const float* __restrict__ A,
                             const float* __restrict__ B,
                             float* __restrict__ C,
                             int M, int N, int K) {
    const int lane  = threadIdx.x & 31;
    const int l15   = lane & 15;
    const int khalf = (lane >> 4) << 1;          // 0 or 2
    const int tm    = blockIdx.x;
    const int tn    = blockIdx.y;

    const float* __restrict__ Arow = A + (size_t)(tm * 16 + l15) * K;
    const int n0 = tn * 64 + l15;

    v8f acc0 = {}, acc1 = {}, acc2 = {}, acc3 = {};

    for (int k = 0; k < K; k += 4) {
        v2f a;
        a.x = Arow[k + khalf];
        a.y = Arow[k + khalf + 1];

        const float* __restrict__ B0 = B + (size_t)(k + khalf) * N;
        const float* __restrict__ B1 = B0 + N;

        v2f b0; b0.x = B0[n0];      b0.y = B1[n0];
        v2f b1; b1.x = B0[n0 + 16]; b1.y = B1[n0 + 16];
        v2f b2; b2.x = B0[n0 + 32]; b2.y = B1[n0 + 32];
        v2f b3; b3.x = B0[n0 + 48]; b3.y = B1[n0 + 48];

        acc0 = WMMA_F32X4(a, b0, acc0);
        acc1 = WMMA_F32X4(a, b1, acc1);
        acc2 = WMMA_F32X4(a, b2, acc2);
        acc3 = WMMA_F32X4(a, b3, acc3);
    }

    const int mbase = tm * 16 + ((lane >> 4) << 3);   // +8 for upper half-wave
    const int nc    = tn * 64 + l15;
#pragma unroll
    for (int r = 0; r < 8; ++r) {
        float* __restrict__ Crow = C + (size_t)(mbase + r) * N + nc;
        Crow[0]  = acc0[r];
        Crow[16] = acc1[r];
        Crow[32] = acc2[r];
        Crow[48] = acc3[r];
    }
}

// ---------------------------------------------------------------------------
// C[N x N] = sigmoid(H @ H^T), H is [N x K] row-major (decode step).
// One wave per 32(M) x 64(N) block: 2 A fragments x 4 B fragments = 8
// independent WMMA chains per K-step; every B fragment reused twice.
// B fragment element B[k][n] = H[n*K + k]: contiguous in k -> clean b64 loads.
// Output written with non-temporal stores (streaming, bypass L2 retention).
// grid = (N/32, N/64), block = 32.
// ---------------------------------------------------------------------------
__global__ void wmma_gemm_nt_sigmoid(const float* __restrict__ H,
                                     float* __restrict__ C,
                                     int N, int K) {
    const int lane  = threadIdx.x & 31;
    const int l15   = lane & 15;
    const int khalf = (lane >> 4) << 1;          // 0 or 2
    const int tm    = blockIdx.x;
    const int tn    = blockIdx.y;

    const float* __restrict__ Arow0 = H + (size_t)(tm * 32 + l15)      * K;
    const float* __restrict__ Arow1 = H + (size_t)(tm * 32 + 16 + l15) * K;
    const int n0 = tn * 64 + l15;
    const float* __restrict__ Br0 = H + (size_t)(n0)      * K;
    const float* __restrict__ Br1 = H + (size_t)(n0 + 16) * K;
    const float* __restrict__ Br2 = H + (size_t)(n0 + 32) * K;
    const float* __restrict__ Br3 = H + (size_t)(n0 + 48) * K;

    v8f acc00 = {}, acc01 = {}, acc02 = {}, acc03 = {};
    v8f acc10 = {}, acc11 = {}, acc12 = {}, acc13 = {};

    for (int k = 0; k < K; k += 4) {
        const int kk = k + khalf;
        v2f aA; aA.x = Arow0[kk]; aA.y = Arow0[kk + 1];
        v2f aB; aB.x = Arow1[kk]; aB.y = Arow1[kk + 1];
        v2f b0; b0.x = Br0[kk];   b0.y = Br0[kk + 1];
        v2f b1; b1.x = Br1[kk];   b1.y = Br1[kk + 1];
        v2f b2; b2.x = Br2[kk];   b2.y = Br2[kk + 1];
        v2f b3; b3.x = Br3[kk];   b3.y = Br3[kk + 1];

        acc00 = WMMA_F32X4(aA, b0, acc00);
        acc01 = WMMA_F32X4(aA, b1, acc01);
        acc02 = WMMA_F32X4(aA, b2, acc02);
        acc03 = WMMA_F32X4(aA, b3, acc03);
        acc10 = WMMA_F32X4(aB, b0, acc10);
        acc11 = WMMA_F32X4(aB, b1, acc11);
        acc12 = WMMA_F32X4(aB, b2, acc12);
        acc13 = WMMA_F32X4(aB, b3, acc13);
    }

    const int mbase0 = tm * 32 + ((lane >> 4) << 3);       // first 16-row tile
    const int mbase1 = mbase0 + 16;                        // second 16-row tile
    const int nc     = tn * 64 + l15;
#pragma unroll
    for (int r = 0; r < 8; ++r) {
        float* __restrict__ Crow0 = C + (size_t)(mbase0 + r) * N + nc;
        float* __restrict__ Crow1 = C + (size_t)(mbase1 + r) * N + nc;
        __builtin_nontemporal_store(sigmoidf_dev(acc00[r]), Crow0 + 0);
        __builtin_nontemporal_store(sigmoidf_dev(acc01[r]), Crow0 + 16);
        __builtin_nontemporal_store(sigmoidf_dev(acc02[r]), Crow0 + 32);
        __builtin_nontemporal_store(sigmoidf_dev(acc03[r]), Crow0 + 48);
        __builtin_nontemporal_store(sigmoidf_dev(acc10[r]), Crow1 + 0);
        __builtin_nontemporal_store(sigmoidf_dev(acc11[r]), Crow1 + 16);
        __builtin_nontemporal_store(sigmoidf_dev(acc12[r]), Crow1 + 32);
        __builtin_nontemporal_store(sigmoidf_dev(acc13[r]), Crow1 + 48);
    }
}

// ---------------------------------------------------------------------------
// f1[i] = dot(H[i], a0); f2[i] = dot(H[i], a1). One wave per row.
// block = 256 (8 waves) -> 8 rows per block.
// ---------------------------------------------------------------------------
__global__ void row_dot2(const float* __restrict__ H,
                         const float* __restrict__ a0,
                         const float* __restrict__ a1,
                         float* __restrict__ f1,
                         float* __restrict__ f2,
                         int D) {
    const int lane = threadIdx.x & 31;
    const int row  = blockIdx.x * 8 + (threadIdx.x >> 5);
    const float* __restrict__ Hrow = H + (size_t)row * D;

    float s1 = 0.0f, s2 = 0.0f;
    for (int d = lane; d < D; d += 32) {
        const float h = Hrow[d];
        s1 += h * a0[d];
        s2 += h * a1[d];
    }
#pragma unroll
    for (int off = 16; off > 0; off >>= 1) {
        s1 += __shfl_xor(s1, off);
        s2 += __shfl_xor(s2, off);
    }
    if (lane == 0) {
        f1[row] = s1;
        f2[row] = s2;
    }
}

// ---------------------------------------------------------------------------
// Per-edge attention: exp_edge[e] = exp(sigmoid(values[e] * (f1[row]+f2[col])))
// and row_sum[row] += exp_edge[e] (atomic). One thread per edge.
// ---------------------------------------------------------------------------
__global__ void edge_logits(const float* __restrict__ f1,
                            const float* __restrict__ f2,
                            const float* __restrict__ vals,
                            const long long* __restrict__ row,
                            const long long* __restrict__ col,
                            float* __restrict__ exp_edge,
                            float* __restrict__ row_sum,
                            int E) {
    const int e = blockIdx.x * blockDim.x + threadIdx.x;
    if (e >= E) return;
    const int r = (int)row[e];
    const int c = (int)col[e];
    const float att = sigmoidf_dev(vals[e] * (f1[r] + f2[c]));
    const float ex  = __expf(att);
    exp_edge[e] = ex;
    atomicAdd(&row_sum[r], ex);
}

// ---------------------------------------------------------------------------
// Sparse aggregation: Hout[r] += (exp_edge[e]/row_sum[r]) * H[c]  (atomic).
// One wave per edge; lanes stride the feature dim.
// block = 256 (8 waves) -> 8 edges per block.
// ---------------------------------------------------------------------------
__global__ void edge_scatter(const float* __restrict__ H,
                             const float* __restrict__ exp_edge,
                             const float* __restrict__ row_sum,
                             const long long* __restrict__ row,
                             const long long* __restrict__ col,
                             float* __restrict__ Hout,
                             int D, int E) {
    const int lane = threadIdx.x & 31;
    const int e    = blockIdx.x * 8 + (threadIdx.x >> 5);
    if (e >= E) return;
    const int r = (int)row[e];
    const int c = (int)col[e];
    const float norm = exp_edge[e] / row_sum[r];
    const float* __restrict__ Hc = H    + (size_t)c * D;
    float*       __restrict__ Hr = Hout + (size_t)r * D;
    for (int d = lane; d < D; d += 32) {
        atomicAdd(&Hr[d], norm * Hc[d]);
    }
}

// ---------------------------------------------------------------------------
// Host-side orchestration
// ---------------------------------------------------------------------------
extern "C" void kernel_launch(void* const* d_in, const int* in_sizes, int n_in,
                              void* d_out, int out_size, void* d_ws, size_t ws_size,
                              hipStream_t stream) {
    (void)in_sizes; (void)n_in; (void)out_size; (void)ws_size;

    constexpr int N    = 16384;
    constexpr int F_IN = 512;
    constexpr int D1   = 512;   // HIDDEN[1]
    constexpr int D2   = 256;   // HIDDEN[2]
    constexpr int E    = 524288;

    const float*     X    = (const float*)d_in[0];
    const float*     W0   = (const float*)d_in[1];
    const float*     W1   = (const float*)d_in[2];
    const float*     a0_0 = (const float*)d_in[3];
    const float*     a1_0 = (const float*)d_in[4];
    const float*     a0_1 = (const float*)d_in[5];
    const float*     a1_1 = (const float*)d_in[6];
    const float*     vals = (const float*)d_in[7];
    const long long* eidx = (const long long*)d_in[8];
    const long long* row  = eidx;
    const long long* col  = eidx + E;

    // Workspace layout (floats)
    float* ws = (float*)d_ws;
    float* Ha       = ws;                               // 16384*512
    float* Hb       = Ha + (size_t)N * D1;              // 16384*512
    float* f1       = Hb + (size_t)N * D1;              // 16384
    float* f2       = f1 + N;                           // 16384
    float* row_sum  = f2 + N;                           // 16384
    float* exp_edge = row_sum + N;                      // E

    float* A_pred = (float*)d_out;

    const dim3 blk32(32, 1, 1);
    const dim3 blk256(256, 1, 1);

    // --- H1 = X @ W0 : [16384 x 512] ---
    wmma_gemm_nn<<<dim3(N / 16, D1 / 64, 1), blk32, 0, stream>>>(X, W0, Ha, N, D1, F_IN);

    // --- GAT layer 1 (D=512): Ha -> Hb ---
    row_dot2<<<dim3(N / 8, 1, 1), blk256, 0, stream>>>(Ha, a0_0, a1_0, f1, f2, D1);
    zero_f32<<<dim3((N + 255) / 256, 1, 1), blk256, 0, stream>>>(row_sum, N);
    edge_logits<<<dim3(E / 256, 1, 1), blk256, 0, stream>>>(f1, f2, vals, row, col,
                                                            exp_edge, row_sum, E);
    zero_f32<<<dim3(((long)N * D1 + 255) / 256, 1, 1), blk256, 0, stream>>>(Hb, (long)N * D1);
    edge_scatter<<<dim3(E / 8, 1, 1), blk256, 0, stream>>>(Ha, exp_edge, row_sum, row, col,
                                                           Hb, D1, E);

    // --- H2 = Hb @ W1 : [16384 x 256] (into Ha) ---
    wmma_gemm_nn<<<dim3(N / 16, D2 / 64, 1), blk32, 0, stream>>>(Hb, W1, Ha, N, D2, D1);

    // --- GAT layer 2 (D=256): Ha -> Hb ---
    row_dot2<<<dim3(N / 8, 1, 1), blk256, 0, stream>>>(Ha, a0_1, a1_1, f1, f2, D2);
    zero_f32<<<dim3((N + 255) / 256, 1, 1), blk256, 0, stream>>>(row_sum, N);
    edge_logits<<<dim3(E / 256, 1, 1), blk256, 0, stream>>>(f1, f2, vals, row, col,
                                                            exp_edge, row_sum, E);
    zero_f32<<<dim3(((long)N * D2 + 255) / 256, 1, 1), blk256, 0, stream>>>(Hb, (long)N * D2);
    edge_scatter<<<dim3(E / 8, 1, 1), blk256, 0, stream>>>(Ha, exp_edge, row_sum, row, col,
                                                           Hb, D2, E);

    // --- A_pred = sigmoid(Hb @ Hb^T) : [16384 x 16384] ---
    wmma_gemm_nt_sigmoid<<<dim3(N / 32, N / 64, 1), blk32, 0, stream>>>(Hb, A_pred, N, D2);
}